// DeepIRT_38611755991713
// MI455X (gfx1250) — compile-verified
//
#include <hip/hip_runtime.h>
#include <hip/hip_bf16.h>
#include <math.h>

// ---------------------------------------------------------------------------
// DeepIRT / DKVMN forward for MI455X (gfx1250, wave32, WMMA f32 16x16x4)
// B=128, L=200, NUM_C=10000, D=128, M=64
// ---------------------------------------------------------------------------

#define BB 128
#define LL 200
#define DD 128
#define MM 64
#define NROWS (BB * LL)          // 25600
#define ROWTILES (NROWS / 16)    // 1600
#define WAVES_PER_BLOCK 8

typedef __attribute__((ext_vector_type(2))) float v2f;
typedef __attribute__((ext_vector_type(8))) float v8f;

__device__ __forceinline__ float sigmoidf_(float x) {
    return 1.0f / (1.0f + __expf(-x));
}

// ---------------------------------------------------------------------------
// K0: k = k_emb[q]   (gather, one thread per element)
// ---------------------------------------------------------------------------
__global__ __launch_bounds__(256) void k0_gather(const int* __restrict__ q,
                                                 const float* __restrict__ k_emb,
                                                 float* __restrict__ k_buf) {
    int idx = blockIdx.x * 256 + threadIdx.x;   // 0 .. B*L*D-1 (exact multiple)
    int row = idx >> 7;                          // /128
    int d   = idx & 127;
    k_buf[idx] = k_emb[(size_t)q[row] * DD + d];
}

// ---------------------------------------------------------------------------
// K1: per-wave 16-row tile:
//   w_logits(16x64) = k_tile * Mk^T          (4 WMMA tiles)
//   e(16x128) = sigmoid(v_tile * eW^T + eb)  (8 WMMA tiles), v = k + v_emb[r]
//   a(16x128) = tanh   (v_tile * aW^T + ab)  (8 WMMA tiles)
// A frag: lane = m + 16h holds A[m][kk*4 + 2h + j] (float2)
// B frag: lane = n + 16h holds W[n][kk*4 + 2h + j] (float2, weights row-major)
// ---------------------------------------------------------------------------
__global__ __launch_bounds__(256) void k1_proj(const float* __restrict__ k_buf,
                                               const int* __restrict__ r,
                                               const float* __restrict__ v_emb,
                                               const float* __restrict__ Mk,
                                               const float* __restrict__ eW,
                                               const float* __restrict__ eb,
                                               const float* __restrict__ aW,
                                               const float* __restrict__ ab,
                                               float* __restrict__ w_out,
                                               float* __restrict__ e_out,
                                               float* __restrict__ a_out) {
    const int wave = blockIdx.x * WAVES_PER_BLOCK + (threadIdx.x >> 5); // < 1600 exact
    const int lane = threadIdx.x & 31;
    const int h    = lane >> 4;      // 0/1 : K-half
    const int l16  = lane & 15;      // row (A) / col (B)
    const int row0 = wave * 16;

    const float* Arow = k_buf + (size_t)(row0 + l16) * DD;
    const int    ridx = r[row0 + l16];
    const float* Vemb = v_emb + ridx * DD;

    // ---- w logits: k * Mk^T, N = 64 -> 4 tiles --------------------------
    for (int n = 0; n < 4; ++n) {
        v8f acc = {};
        const float* Brow = Mk + (size_t)(n * 16 + l16) * DD;
        for (int kk = 0; kk < 32; ++kk) {
            const int off = kk * 4 + 2 * h;
            v2f af = *(const v2f*)(Arow + off);
            v2f bf = *(const v2f*)(Brow + off);
            acc = __builtin_amdgcn_wmma_f32_16x16x4_f32(
                false, af, false, bf, (short)0, acc, false, false);
        }
        const int cc = n * 16 + l16;
        for (int g = 0; g < 8; ++g) {
            int rr = row0 + g + 8 * h;
            w_out[(size_t)rr * MM + cc] = acc[g];
        }
    }

    // ---- e = sigmoid(v * eW^T + eb), N = 128 -> 8 tiles -----------------
    for (int n = 0; n < 8; ++n) {
        v8f acc = {};
        const float* Brow = eW + (size_t)(n * 16 + l16) * DD;
        for (int kk = 0; kk < 32; ++kk) {
            const int off = kk * 4 + 2 * h;
            v2f af = *(const v2f*)(Arow + off);
            v2f ve = *(const v2f*)(Vemb + off);
            af += ve;                                  // v = k + v_emb[r]
            v2f bf = *(const v2f*)(Brow + off);
            acc = __builtin_amdgcn_wmma_f32_16x16x4_f32(
                false, af, false, bf, (short)0, acc, false, false);
        }
        const int cc = n * 16 + l16;
        const float bias = eb[cc];
        for (int g = 0; g < 8; ++g) {
            int rr = row0 + g + 8 * h;
            e_out[(size_t)rr * DD + cc] = sigmoidf_(acc[g] + bias);
        }
    }

    // ---- a = tanh(v * aW^T + ab), N = 128 -> 8 tiles --------------------
    for (int n = 0; n < 8; ++n) {
        v8f acc = {};
        const float* Brow = aW + (size_t)(n * 16 + l16) * DD;
        for (int kk = 0; kk < 32; ++kk) {
            const int off = kk * 4 + 2 * h;
            v2f af = *(const v2f*)(Arow + off);
            v2f ve = *(const v2f*)(Vemb + off);
            af += ve;
            v2f bf = *(const v2f*)(Brow + off);
            acc = __builtin_amdgcn_wmma_f32_16x16x4_f32(
                false, af, false, bf, (short)0, acc, false, false);
        }
        const int cc = n * 16 + l16;
        const float bias = ab[cc];
        for (int g = 0; g < 8; ++g) {
            int rr = row0 + g + 8 * h;
            a_out[(size_t)rr * DD + cc] = tanhf(acc[g] + bias);
        }
    }
}

// ---------------------------------------------------------------------------
// K2: in-place row softmax over M=64 (one thread per row)
// ---------------------------------------------------------------------------
__global__ __launch_bounds__(256) void k2_softmax(float* __restrict__ w) {
    int row = blockIdx.x * 256 + threadIdx.x;      // 25600 exact
    float* p = w + (size_t)row * MM;
    float mx = -3.4e38f;
    for (int m = 0; m < MM; ++m) mx = fmaxf(mx, p[m]);
    float s = 0.0f;
    for (int m = 0; m < MM; ++m) {
        float ex = __expf(p[m] - mx);
        p[m] = ex;
        s += ex;
    }
    float inv = 1.0f / s;
    for (int m = 0; m < MM; ++m) p[m] *= inv;
}

// ---------------------------------------------------------------------------
// K3: sequential scan over L. One block per batch; Mvt (64x128) in LDS.
//   read[d]  = sum_m w[m]*Mvt[m][d]
//   Mvt[m][d] = Mvt[m][d]*(1 - w[m]*e[d]) + w[m]*a[d]
// ---------------------------------------------------------------------------
__global__ __launch_bounds__(256) void k3_scan(const float* __restrict__ Mv0,
                                               const float* __restrict__ w,
                                               const float* __restrict__ e,
                                               const float* __restrict__ a,
                                               float* __restrict__ reads) {
    __shared__ float Mvt[MM * DD];   // 32 KB
    __shared__ float wt[MM];
    __shared__ float et[DD];
    __shared__ float at[DD];

    const int b   = blockIdx.x;
    const int tid = threadIdx.x;

    for (int i = tid; i < MM * DD; i += 256) Mvt[i] = Mv0[i];
    __syncthreads();

    for (int t = 0; t < LL; ++t) {
        const size_t row = (size_t)b * LL + t;
        if (tid < MM)  wt[tid] = w[row * MM + tid];
        if (tid < DD)  et[tid] = e[row * DD + tid];
        else           at[tid - DD] = a[row * DD + (tid - DD)];
        __syncthreads();

        if (tid < DD) {
            float s = 0.0f;
            #pragma unroll 8
            for (int m = 0; m < MM; ++m) s += wt[m] * Mvt[m * DD + tid];
            reads[row * DD + tid] = s;
        }
        __syncthreads();

        for (int i = tid; i < MM * DD; i += 256) {
            const int m = i >> 7;
            const int d = i & 127;
            const float wm = wt[m];
            Mvt[i] = Mvt[i] * (1.0f - wm * et[d]) + wm * at[d];
        }
        __syncthreads();
    }
}

// ---------------------------------------------------------------------------
// K4: f = tanh([reads||k] * fW^T + fb)  (K=256, 8 WMMA tiles of 16x16 per wave)
//     ability = tanh(f . abW + abb); diff = tanh(k . dfW + dfb)
//     p = sigmoid(3*ability - diff)
// The abW dot is fused into the WMMA epilogue; 16-lane shfl_xor reduction.
// ---------------------------------------------------------------------------
__global__ __launch_bounds__(256) void k4_final(const float* __restrict__ reads,
                                                const float* __restrict__ k_buf,
                                                const float* __restrict__ fW,
                                                const float* __restrict__ fb,
                                                const float* __restrict__ abW,
                                                const float* __restrict__ abb,
                                                const float* __restrict__ dfW,
                                                const float* __restrict__ dfb,
                                                float* __restrict__ out) {
    const int wave = blockIdx.x * WAVES_PER_BLOCK + (threadIdx.x >> 5);
    const int lane = threadIdx.x & 31;
    const int h    = lane >> 4;
    const int l16  = lane & 15;
    const int row0 = wave * 16;

    const float* Rrow = reads + (size_t)(row0 + l16) * DD;
    const float* Krow = k_buf + (size_t)(row0 + l16) * DD;

    float rowacc[8];
    #pragma unroll
    for (int g = 0; g < 8; ++g) rowacc[g] = 0.0f;

    for (int n = 0; n < 8; ++n) {
        v8f acc = {};
        const float* Brow = fW + (size_t)(n * 16 + l16) * (2 * DD);
        // kk < 32  -> concat half 0 (reads); kk >= 32 -> half 1 (k). Wave-uniform.
        for (int kk = 0; kk < 32; ++kk) {
            const int off = kk * 4 + 2 * h;
            v2f af = *(const v2f*)(Rrow + off);
            v2f bf = *(const v2f*)(Brow + off);
            acc = __builtin_amdgcn_wmma_f32_16x16x4_f32(
                false, af, false, bf, (short)0, acc, false, false);
        }
        for (int kk = 32; kk < 64; ++kk) {
            const int off = kk * 4 + 2 * h;
            v2f af = *(const v2f*)(Krow + (off - DD));
            v2f bf = *(const v2f*)(Brow + off);
            acc = __builtin_amdgcn_wmma_f32_16x16x4_f32(
                false, af, false, bf, (short)0, acc, false, false);
        }
        const int cc = n * 16 + l16;
        const float bias = fb[cc];
        const float aw   = abW[cc];
        #pragma unroll
        for (int g = 0; g < 8; ++g)
            rowacc[g] += tanhf(acc[g] + bias) * aw;   // fused f . abW
    }

    // reduce over the 16 lanes of each half-wave (rows g+8h)
    #pragma unroll
    for (int g = 0; g < 8; ++g) {
        rowacc[g] += __shfl_xor(rowacc[g], 1, 32);
        rowacc[g] += __shfl_xor(rowacc[g], 2, 32);
        rowacc[g] += __shfl_xor(rowacc[g], 4, 32);
        rowacc[g] += __shfl_xor(rowacc[g], 8, 32);
    }

    if (l16 < 8) {
        float mysum = 0.0f;
        #pragma unroll
        for (int g = 0; g < 8; ++g)
            if (l16 == g) mysum = rowacc[g];
        const int rr = row0 + l16 + 8 * h;
        const float ability = tanhf(mysum + abb[0]);

        const float* Kr = k_buf + (size_t)rr * DD;
        float dd = 0.0f;
        for (int d = 0; d < DD; ++d) dd += Kr[d] * dfW[d];
        const float diff = tanhf(dd + dfb[0]);

        out[rr] = sigmoidf_(3.0f * ability - diff);
    }
}

// ---------------------------------------------------------------------------
// launch
// ---------------------------------------------------------------------------
extern "C" void kernel_launch(void* const* d_in, const int* in_sizes, int n_in,
                              void* d_out, int out_size, void* d_ws, size_t ws_size,
                              hipStream_t stream) {
    const int*   q     = (const int*)  d_in[0];
    const int*   r     = (const int*)  d_in[1];
    const float* k_emb = (const float*)d_in[2];
    const float* v_emb = (const float*)d_in[3];
    const float* Mk    = (const float*)d_in[4];
    const float* Mv0   = (const float*)d_in[5];
    const float* fW    = (const float*)d_in[6];
    const float* fb    = (const float*)d_in[7];
    const float* eW    = (const float*)d_in[8];
    const float* eb    = (const float*)d_in[9];
    const float* aW    = (const float*)d_in[10];
    const float* ab    = (const float*)d_in[11];
    const float* abW   = (const float*)d_in[12];
    const float* abb   = (const float*)d_in[13];
    const float* dfW   = (const float*)d_in[14];
    const float* dfb   = (const float*)d_in[15];
    float* out = (float*)d_out;

    float* ws    = (float*)d_ws;
    float* k_buf = ws;                        // 25600*128
    float* w_buf = k_buf + (size_t)NROWS * DD;  // 25600*64
    float* e_buf = w_buf + (size_t)NROWS * MM;  // 25600*128
    float* a_buf = e_buf + (size_t)NROWS * DD;  // 25600*128
    float* r_buf = a_buf + (size_t)NROWS * DD;  // 25600*128 (reads)

    k0_gather<<<(NROWS * DD) / 256, 256, 0, stream>>>(q, k_emb, k_buf);
    k1_proj<<<ROWTILES / WAVES_PER_BLOCK, 256, 0, stream>>>(
        k_buf, r, v_emb, Mk, eW, eb, aW, ab, w_buf, e_buf, a_buf);
    k2_softmax<<<NROWS / 256, 256, 0, stream>>>(w_buf);
    k3_scan<<<BB, 256, 0, stream>>>(Mv0, w_buf, e_buf, a_buf, r_buf);
    k4_final<<<ROWTILES / WAVES_PER_BLOCK, 256, 0, stream>>>(
        r_buf, k_buf, fW, fb, abW, abb, dfW, dfb, out);
}